// Router_764504178793
// MI455X (gfx1250) — compile-verified
//
#include <hip/hip_runtime.h>
#include <hip/hip_bf16.h>

typedef float v2f __attribute__((ext_vector_type(2)));
typedef float v4f __attribute__((ext_vector_type(4)));
typedef float v8f __attribute__((ext_vector_type(8)));
typedef int   v4i __attribute__((vector_size(16)));   // matches builtin's pointee type

#define GROUPS  8
#define SEQ     4096
#define DMODEL  1024
#define EXPERTS 64
#define TOKENS  (GROUPS * SEQ)   /* 32768 */
#define CAP     1024

#if defined(__has_builtin)
#if __has_builtin(__builtin_amdgcn_global_load_async_to_lds_b128) && \
    __has_builtin(__builtin_amdgcn_s_wait_asynccnt)
#define HAVE_ASYNC_COPY 1
#endif
#endif

// ---------------------------------------------------------------------------
// Kernel 0: one-time 256 KB transpose gate_kernel [D,E] -> gkT [E,D] so that
// WMMA B-fragments (two K-adjacent elements per lane) become single b64 loads.
// ---------------------------------------------------------------------------
__global__ __launch_bounds__(256)
void transpose_gate(const float* __restrict__ gk, float* __restrict__ gkT)
{
    const int e = blockIdx.x;                       // 64 blocks
    for (int d = threadIdx.x; d < DMODEL; d += 256)
        gkT[(size_t)e * DMODEL + d] = gk[(size_t)d * EXPERTS + e];
}

// ---------------------------------------------------------------------------
// Kernel 1: logits = x @ gate_kernel + bias; probs = softmax(logits/temp);
// store transposed probsT[E][T]. One wave owns a 32-token x 64-expert tile:
// 8 fp32 WMMA accumulators, B fragments reused across the two 16-token tiles.
// Per K-step: 2 b64 (A) + 4 b64 (B) + 8x v_wmma_f32_16x16x4_f32.
// ---------------------------------------------------------------------------
__global__ __launch_bounds__(256)
void router_gemm_softmax(const float* __restrict__ x,
                         const float* __restrict__ gkT,  // [E, D]
                         const float* __restrict__ gb,   // [E]
                         const float* __restrict__ temp, // [1]
                         float* __restrict__ probsT)     // [E, T]
{
    __shared__ float lds[8][32 * 65];   // per-wave 32x64 tile, pad 65 = conflict-free

    const int lane = threadIdx.x & 31;
    const int wave = threadIdx.x >> 5;
    const int tile = blockIdx.x * 8 + wave;  // 32-token tile id
    const int t0   = tile * 32;

    const int half = lane >> 4;        // low/high half-wave
    const int l16  = lane & 15;

    // A fragment (16x4 f32): lanes 0-15 hold K=k,k+1 of row M=lane;
    // lanes 16-31 hold K=k+2,k+3 of row M=lane-16.
    const float* arow0 = x + (size_t)(t0 + l16) * DMODEL + 2 * half;
    const float* arow1 = arow0 + (size_t)16 * DMODEL;
    // B fragment (4x16 f32) from gkT: lane needs gkT[n][kb], gkT[n][kb+1].
    const float* brow  = gkT + (size_t)l16 * DMODEL + 2 * half;

    v8f acc[2][4];
#pragma unroll
    for (int m = 0; m < 2; ++m)
#pragma unroll
        for (int nt = 0; nt < 4; ++nt) acc[m][nt] = (v8f)0.0f;

    for (int k = 0; k < DMODEL; k += 4) {
        const v2f a0 = *(const v2f*)(arow0 + k);
        const v2f a1 = *(const v2f*)(arow1 + k);
#pragma unroll
        for (int nt = 0; nt < 4; ++nt) {
            const v2f b = *(const v2f*)(brow + (size_t)nt * 16 * DMODEL + k);
            acc[0][nt] = __builtin_amdgcn_wmma_f32_16x16x4_f32(
                false, a0, false, b, (short)0, acc[0][nt], false, false);
            acc[1][nt] = __builtin_amdgcn_wmma_f32_16x16x4_f32(
                false, a1, false, b, (short)0, acc[1][nt], false, false);
        }
    }

    const float invT = 1.0f / fmaxf(temp[0], 0.1f);
    float bias[4];
#pragma unroll
    for (int nt = 0; nt < 4; ++nt) bias[nt] = gb[nt * 16 + l16];

    // C/D layout: acc[m][nt] vgpr r -> (tok = m*16 + r + 8*half, e = nt*16 + l16).
    // A token's 64 logits live in one 16-lane group: width-16 shuffle butterflies.
    float* my = &lds[wave][0];
#pragma unroll
    for (int m = 0; m < 2; ++m) {
#pragma unroll
        for (int r = 0; r < 8; ++r) {
            float v[4];
#pragma unroll
            for (int nt = 0; nt < 4; ++nt) v[nt] = (acc[m][nt][r] + bias[nt]) * invT;

            float mx = fmaxf(fmaxf(v[0], v[1]), fmaxf(v[2], v[3]));
#pragma unroll
            for (int s = 1; s < 16; s <<= 1) mx = fmaxf(mx, __shfl_xor(mx, s, 16));

            float p[4];
            float sum = 0.0f;
#pragma unroll
            for (int nt = 0; nt < 4; ++nt) { p[nt] = __expf(v[nt] - mx); sum += p[nt]; }
#pragma unroll
            for (int s = 1; s < 16; s <<= 1) sum += __shfl_xor(sum, s, 16);

            const float rinv = 1.0f / sum;
            const int tok = m * 16 + r + 8 * half;
#pragma unroll
            for (int nt = 0; nt < 4; ++nt)
                my[tok * 65 + nt * 16 + l16] = p[nt] * rinv;
        }
    }
    __syncthreads();

    // Coalesced transposed store: each iteration writes 2 experts x 16 tokens.
    const int tt = lane & 15;
    const int eh = lane >> 4;
    for (int i = 0; i < 64; ++i) {
        const int e  = (i & 31) * 2 + eh;
        const int th = i >> 5;
        probsT[(size_t)e * TOKENS + t0 + th * 16 + tt] = my[(th * 16 + tt) * 65 + e];
    }
}

// ---------------------------------------------------------------------------
// Kernel 2: per-expert exact top-1024 via 4-pass radix select.
// One workgroup per expert; the full 128 KB row staged in LDS (CDNA5: 320 KB
// per workgroup) with async copy-to-LDS when available. Ties are taken in
// ascending token-index order (jax.lax.top_k semantics) via a block scan.
// ---------------------------------------------------------------------------
__global__ __launch_bounds__(1024)
void expert_topk_select(const float* __restrict__ probsT,
                        float* __restrict__ outMask,  // [E*T] 0/1
                        float* __restrict__ outW,     // [E*T]
                        float* __restrict__ outLoss)  // [1]
{
    __shared__ __align__(16) float vals[TOKENS];   // 128 KB
    __shared__ unsigned hist[256];
    __shared__ unsigned sc[1024];
    __shared__ unsigned s_prefix, s_K;

    const int e   = blockIdx.x;
    const int tid = threadIdx.x;
    const float* row = probsT + (size_t)e * TOKENS;

#if defined(HAVE_ASYNC_COPY)
    for (int i = tid; i < TOKENS / 4; i += 1024)
        __builtin_amdgcn_global_load_async_to_lds_b128(
            (__attribute__((address_space(1))) v4i*)(row + i * 4),
            (__attribute__((address_space(3))) v4i*)(vals + i * 4),
            0, 0);
    __builtin_amdgcn_s_wait_asynccnt(0);
#else
    for (int i = tid; i < TOKENS / 4; i += 1024)
        ((v4f*)vals)[i] = ((const v4f*)row)[i];
#endif
    if (tid == 0) { s_prefix = 0u; s_K = CAP; }
    __syncthreads();

    // Radix select: softmax probs are positive -> uint bit order == float order.
    for (int pass = 0; pass < 4; ++pass) {
        if (tid < 256) hist[tid] = 0u;
        __syncthreads();
        const unsigned pref  = s_prefix;
        const int      shift = 24 - pass * 8;
        for (int i = tid; i < TOKENS / 4; i += 1024) {
            const v4f v = ((const v4f*)vals)[i];
#pragma unroll
            for (int c = 0; c < 4; ++c) {
                const unsigned u = __float_as_uint(v[c]);
                const bool cand = (pass == 0) || ((u >> (shift + 8)) == pref);
                if (cand) atomicAdd(&hist[(u >> shift) & 0xFFu], 1u);
            }
        }
        __syncthreads();
        if (tid == 0) {
            unsigned K = s_K, cum = 0u, d = 0u;
            for (int b = 255; b >= 0; --b) {
                const unsigned h = hist[b];
                if (cum + h >= K) { d = (unsigned)b; break; }
                cum += h;
            }
            s_prefix = (pref << 8) | d;
            s_K      = K - cum;       // ties (== kth value) still to take
        }
        __syncthreads();
    }
    const unsigned thr     = s_prefix;  // exact bits of the 1024th-largest prob
    const unsigned numTies = s_K;

    // Index-ordered tie ranks: contiguous 32-token chunk per thread + block scan.
    const int base = tid * 32;
    unsigned cnt = 0u;
#pragma unroll
    for (int q = 0; q < 8; ++q) {
        const v4f v = ((const v4f*)vals)[(base >> 2) + q];
#pragma unroll
        for (int c = 0; c < 4; ++c)
            cnt += (__float_as_uint(v[c]) == thr) ? 1u : 0u;
    }
    sc[tid] = cnt;
    __syncthreads();
    for (int off = 1; off < 1024; off <<= 1) {
        const unsigned v = (tid >= off) ? sc[tid - off] : 0u;
        __syncthreads();
        sc[tid] += v;
        __syncthreads();
    }
    unsigned rank = sc[tid] - cnt;  // equal-valued tokens before my chunk

    v4f* maskOut = (v4f*)(outMask + (size_t)e * TOKENS + base);
    v4f* wOut    = (v4f*)(outW    + (size_t)e * TOKENS + base);
#pragma unroll
    for (int q = 0; q < 8; ++q) {
        const v4f v = ((const v4f*)vals)[(base >> 2) + q];
        v4f m4, w4;
#pragma unroll
        for (int c = 0; c < 4; ++c) {
            const unsigned u = __float_as_uint(v[c]);
            bool sel;
            if (u > thr)        sel = true;
            else if (u == thr) { sel = (rank < numTies); ++rank; }
            else                sel = false;
            m4[c] = sel ? 1.0f : 0.0f;
            w4[c] = sel ? v[c] : 0.0f;
        }
        maskOut[q] = m4;
        wOut[q]    = w4;
    }
    if (e == 0 && tid == 0) outLoss[0] = 0.0f;
}

// ---------------------------------------------------------------------------
extern "C" void kernel_launch(void* const* d_in, const int* in_sizes, int n_in,
                              void* d_out, int out_size, void* d_ws, size_t ws_size,
                              hipStream_t stream)
{
    const float* x    = (const float*)d_in[0];  // [G,S,D]
    const float* gk   = (const float*)d_in[1];  // [D,E]
    const float* gb   = (const float*)d_in[2];  // [E]
    const float* temp = (const float*)d_in[3];  // [1]
    // d_in[4] = expert_capacity (compile-time constant CAP=1024)

    float* out     = (float*)d_out;
    float* outMask = out;                                   // E*G*S
    float* outW    = out + (size_t)EXPERTS * TOKENS;        // E*G*S
    float* outLoss = out + 2 * (size_t)EXPERTS * TOKENS;    // 1

    float* probsT = (float*)d_ws;                           // 8 MB
    float* gkT    = probsT + (size_t)EXPERTS * TOKENS;      // +256 KB

    transpose_gate<<<EXPERTS, 256, 0, stream>>>(gk, gkT);
    router_gemm_softmax<<<TOKENS / 32 / 8, 256, 0, stream>>>(x, gkT, gb, temp, probsT);
    expert_topk_select<<<EXPERTS, 1024, 0, stream>>>(probsT, outMask, outW, outLoss);
}